// PowerMethod_19928648254205
// MI455X (gfx1250) — compile-verified
//
#include <hip/hip_runtime.h>
#include <hip/hip_bf16.h>
#include <stdint.h>

// COO power iteration: v_{k+1}[row] += w * v_k[col], K=8, D=64.
// Memory-bound (0.25 FLOP/B). Working set (~96 MB) fits MI455X 192 MB L2.
// One wave32 per edge, float2 per lane, hardware f32 atomics for scatter.
// Edge lists staged into LDS via gfx1250 async global->LDS (ASYNCcnt path).

#define EDGES_PER_BLOCK 256
#define BLOCK_THREADS   256
#define DIM             64

__global__ __launch_bounds__(BLOCK_THREADS) void
spmv_scatter_iter(const float* __restrict__ vin,
                  float* __restrict__ vout,
                  const int* __restrict__ edge_row,
                  const int* __restrict__ edge_col,
                  const float* __restrict__ edge_w,
                  int n_edges) {
    __shared__ int   s_row[EDGES_PER_BLOCK];
    __shared__ int   s_col[EDGES_PER_BLOCK];
    __shared__ float s_w[EDGES_PER_BLOCK];

    const int t          = threadIdx.x;
    const int blockStart = blockIdx.x * EDGES_PER_BLOCK;
    const int count      = min(EDGES_PER_BLOCK, n_edges - blockStart);

    // ---- Stage this block's edge data into LDS via async global->LDS ----
    if (t < count) {
        const int g = blockStart + t;
        uint32_t aR = (uint32_t)(uintptr_t)&s_row[t];
        uint32_t aC = (uint32_t)(uintptr_t)&s_col[t];
        uint32_t aW = (uint32_t)(uintptr_t)&s_w[t];
        const int*   pR = edge_row + g;
        const int*   pC = edge_col + g;
        const float* pW = edge_w + g;
        asm volatile("global_load_async_to_lds_b32 %0, %1, off"
                     :: "v"(aR), "v"(pR) : "memory");
        asm volatile("global_load_async_to_lds_b32 %0, %1, off"
                     :: "v"(aC), "v"(pC) : "memory");
        asm volatile("global_load_async_to_lds_b32 %0, %1, off"
                     :: "v"(aW), "v"(pW) : "memory");
    }
    // Wait for this wave's async transfers, then workgroup barrier so every
    // wave sees the full staged chunk.
    asm volatile("s_wait_asynccnt 0x0" ::: "memory");
    __syncthreads();

    const int lane = t & 31;   // wave32
    const int wv   = t >> 5;   // 8 waves per block

    // Each wave processes 32 consecutive edges of the staged chunk.
    #pragma unroll 4
    for (int i = 0; i < 32; ++i) {
        const int e = (wv << 5) + i;
        if (blockStart + e >= n_edges) break;

        const int   row = s_row[e];   // broadcast ds_load (same addr all lanes)
        const int   col = s_col[e];
        const float wt  = s_w[e];

        // Gather: one coalesced 256B line per edge (float2 per lane).
        const float2 x =
            *reinterpret_cast<const float2*>(vin + (col << 6) + (lane << 1));

        // Scatter-add: hardware global_atomic_add_f32 (L2 atomic units).
        float* dst = vout + (row << 6) + (lane << 1);
        unsafeAtomicAdd(dst,     x.x * wt);
        unsafeAtomicAdd(dst + 1, x.y * wt);
    }
}

extern "C" void kernel_launch(void* const* d_in, const int* in_sizes, int n_in,
                              void* d_out, int out_size, void* d_ws, size_t ws_size,
                              hipStream_t stream) {
    const float* v0  = (const float*)d_in[0];
    const int*   er  = (const int*)d_in[1];
    const int*   ec  = (const int*)d_in[2];
    const float* ew  = (const float*)d_in[3];
    float* out = (float*)d_out;
    float* buf = (float*)d_ws;   // ping buffer (needs N*DIM floats)

    const int n_edges = in_sizes[1];
    const int n_nodes = out_size / DIM;
    const size_t vbytes = (size_t)n_nodes * DIM * sizeof(float);

    const int nBlocks = (n_edges + EDGES_PER_BLOCK - 1) / EDGES_PER_BLOCK;

    // K = 8 iterations, ping-pong ws/out so the final result lands in d_out.
    const float* src = v0;
    for (int k = 0; k < 8; ++k) {
        float* dst = (k & 1) ? out : buf;
        hipMemsetAsync(dst, 0, vbytes, stream);
        spmv_scatter_iter<<<nBlocks, BLOCK_THREADS, 0, stream>>>(
            src, dst, er, ec, ew, n_edges);
        src = dst;
    }
}